// DotMaskLayer_16647293239324
// MI455X (gfx1250) — compile-verified
//
#include <hip/hip_runtime.h>
#include <cmath>

// ---------------------------------------------------------------------------
// DotMaskLayer on MI455X (gfx1250, wave32).
//
// Roofline: 1.08 GB streamed (AUX) vs 0.54 GFLOP -> HBM-bound (~46 us floor
// at 23.3 TB/s). Strategy: one batch row per wave; stage the masked Ym vector
// (257 f32) in LDS; run the j-reduction on V_WMMA_F32_16X16X4_F32 with the
// A-matrix broadcast across all 16 M rows (redundant M is free under the
// bandwidth roofline and keeps VALU for addressing only). B-tiles are read
// straight from HBM, each AUX byte exactly once, coalesced.
// ---------------------------------------------------------------------------

typedef float v2f __attribute__((ext_vector_type(2)));
typedef float v8f __attribute__((ext_vector_type(8)));

#define WAVES_PER_BLOCK 8
#define YM_STRIDE 260            // 257 rounded up, keeps 8B alignment per wave
#define D_DIM 256
#define H_DIM 64
#define AUX_ROW 16448            // (1+256)*64

__global__ __launch_bounds__(256, 2)
void DotMaskLayer_kernel(const float* __restrict__ X,
                         const float* __restrict__ AUX,
                         const int* __restrict__ K,
                         float* __restrict__ OUT)
{
    __shared__ float ym[WAVES_PER_BLOCK * YM_STRIDE];

    const int tid  = threadIdx.x;
    const int lane = tid & 31;
    const int wave = __builtin_amdgcn_readfirstlane(tid >> 5);
    const int b    = blockIdx.x * WAVES_PER_BLOCK + wave;   // wave-uniform

    const int k = K[b];                                     // wave-uniform -> SMEM
    const float* __restrict__ xrow = X   + (long)b * D_DIM;
    const float* __restrict__ wrow = AUX + (long)b * AUX_ROW;
    float* yw = &ym[wave * YM_STRIDE];

    // ---- stage masked Ym into LDS: Ym[j] = (X[b,j] or 1) * keep(j,k) ------
    #pragma unroll
    for (int i = 0; i < 8; ++i) {
        int j = lane + 32 * i;
        float x = xrow[j];
        bool keep = (j < k) | ((j >= 128) & (j < 128 + k));
        yw[j] = keep ? x : 0.0f;
    }
    if (lane == 0) {
        // bias element j=256: keep iff (256 < k) | (256 < 128+k)  <=>  k > 128
        yw[256] = (k > 128) ? 1.0f : 0.0f;
    }
    __syncthreads();

    // ---- WMMA GEMV: D = A(16x4, Ym broadcast over M) x B(4x16, W tile) ----
    const int hlo = lane & 15;                 // column N of the B operand
    const int kb  = (lane < 16) ? 0 : 2;       // K rows covered by this lane half

    // A operand source: lanes 0-15 read {Ym[j0],Ym[j0+1]}, lanes 16-31 read
    // {Ym[j0+2],Ym[j0+3]} -> broadcast ds_load_b64 (8B aligned: 16*s (+8)).
    const float* ya = yw + ((lane < 16) ? 0 : 2);

    // B operand source for h-tile 0; tiles 1..3 are +16/+32/+48 floats,
    // next K row is +64 floats, next step is +256 floats.
    const float* wp = wrow + kb * H_DIM + hlo;

    v8f acc0 = {}, acc1 = {}, acc2 = {}, acc3 = {};

    #pragma unroll 2
    for (int s = 0; s < 64; ++s) {
        v2f a = *(const v2f*)(ya + 4 * s);

        v2f bv0 = { wp[0],  wp[64]  };
        v2f bv1 = { wp[16], wp[80]  };
        v2f bv2 = { wp[32], wp[96]  };
        v2f bv3 = { wp[48], wp[112] };
        wp += 4 * H_DIM;

        acc0 = __builtin_amdgcn_wmma_f32_16x16x4_f32(false, a, false, bv0,
                                                     (short)0, acc0, false, false);
        acc1 = __builtin_amdgcn_wmma_f32_16x16x4_f32(false, a, false, bv1,
                                                     (short)0, acc1, false, false);
        acc2 = __builtin_amdgcn_wmma_f32_16x16x4_f32(false, a, false, bv2,
                                                     (short)0, acc2, false, false);
        acc3 = __builtin_amdgcn_wmma_f32_16x16x4_f32(false, a, false, bv3,
                                                     (short)0, acc3, false, false);
    }

    // ---- bias row (j = 256), tanh, store. Row M=0 of D lives in VGPR 0,
    // lanes 0-15 (N = 0..15); all M rows are identical so element 0 is dot[h].
    float ym256 = yw[256];
    if (lane < 16) {
        const float* wb = wrow + 256 * H_DIM + hlo;
        float d0 = acc0[0] + ym256 * wb[0];
        float d1 = acc1[0] + ym256 * wb[16];
        float d2 = acc2[0] + ym256 * wb[32];
        float d3 = acc3[0] + ym256 * wb[48];
        float* o = OUT + (long)b * H_DIM + hlo;
        o[0]  = tanhf(d0);
        o[16] = tanhf(d1);
        o[32] = tanhf(d2);
        o[48] = tanhf(d3);
    }
}

extern "C" void kernel_launch(void* const* d_in, const int* in_sizes, int n_in,
                              void* d_out, int out_size, void* d_ws, size_t ws_size,
                              hipStream_t stream) {
    const float* X   = (const float*)d_in[0];
    const float* AUX = (const float*)d_in[1];
    const int*   K   = (const int*)d_in[2];
    float* OUT = (float*)d_out;

    const int B = in_sizes[2];                 // 16384 rows
    dim3 grid(B / WAVES_PER_BLOCK), block(256);
    DotMaskLayer_kernel<<<grid, block, 0, stream>>>(X, AUX, K, OUT);
}